// Loss_47278999995112
// MI455X (gfx1250) — compile-verified
//
#include <hip/hip_runtime.h>

// Problem constants from the reference: B=2, S=48, T=16, fp32.
#define SDIM 48
#define TDIM 16
#define NBATCH 2
#define NPOINTS (NBATCH * SDIM * SDIM * SDIM)   // 221184
#define NTOT (NPOINTS * TDIM)                   // 3538944
#define NCHUNK (NTOT / 4)                       // 884736 float4 chunks

#define GRID_A 512
#define GRID_B 1024

typedef __attribute__((ext_vector_type(2))) float v2f;
typedef __attribute__((ext_vector_type(4))) float v4f;
typedef __attribute__((ext_vector_type(8))) float v8f;

// ---------------- branchless torch.gradient helpers ----------------
// First derivative along t for a 4-wide chunk starting at t0 (pc -> element t0).
__device__ __forceinline__ v4f tderiv(const float* __restrict__ pc, int t0, const v4f c) {
    const float lv = pc[max(t0 - 1, 0) - t0];            // pc[-1], or pc[0] at t0==0
    const float rv = pc[min(t0 + 4, TDIM - 1) - t0];     // pc[4], or pc[3] at the end
    const float w0 = (t0 == 0) ? 1.0f : 0.5f;
    const float w3 = (t0 + 4 == TDIM) ? 1.0f : 0.5f;
    v4f d;
    d[0] = (c[1] - lv) * w0;
    d[1] = (c[2] - c[0]) * 0.5f;
    d[2] = (c[3] - c[1]) * 0.5f;
    d[3] = (rv - c[2]) * w3;
    return d;
}

// First derivative only (for P), along a spatial axis: stride s, coord i.
__device__ __forceinline__ v4f sderiv1(const float* __restrict__ pc, int i, int s) {
    const int om1 = (max(i - 1, 0) - i) * s;
    const int op1 = (min(i + 1, SDIM - 1) - i) * s;
    const float wc = (i > 0 && i < SDIM - 1) ? 0.5f : 1.0f;
    const v4f fm1 = *(const v4f*)(pc + om1);
    const v4f fp1 = *(const v4f*)(pc + op1);
    return (fp1 - fm1) * wc;
}

// First + composed-second derivative (grad(grad)) along a spatial axis.
// pc points at the center element (coordinate i); c is its value (already loaded).
// Branchless, exact match of _grad(_grad(x)) including one-sided edges:
//   gl = g(max(i-1,0)) = ((i==0 ? f_p1 : f_0) - f_m2) * wl,  wl = (i>1)   ? .5 : 1
//   gr = g(min(i+1,n-1)) = (f_p2 - (i==n-1 ? f_m1 : f_0)) * wr, wr = (i<n-2) ? .5 : 1
//   d2 = (gr - gl) * wc,  d1 = (f_p1 - f_m1) * wc,  wc = (0<i<n-1) ? .5 : 1
__device__ __forceinline__ void sderiv12(const float* __restrict__ pc, int i, int s,
                                         const v4f c, v4f& d1o, v4f& d2o) {
    const int n = SDIM;
    const int om2 = (max(i - 2, 0) - i) * s;
    const int om1 = (max(i - 1, 0) - i) * s;
    const int op1 = (min(i + 1, n - 1) - i) * s;
    const int op2 = (min(i + 2, n - 1) - i) * s;
    const v4f fm2 = *(const v4f*)(pc + om2);
    const v4f fm1 = *(const v4f*)(pc + om1);
    const v4f fp1 = *(const v4f*)(pc + op1);
    const v4f fp2 = *(const v4f*)(pc + op2);
    const float wc = (i > 0 && i < n - 1) ? 0.5f : 1.0f;
    const float wl = (i > 1) ? 0.5f : 1.0f;
    const float wr = (i < n - 2) ? 0.5f : 1.0f;
    const v4f al = (i == 0) ? fp1 : c;
    const v4f ar = (i == n - 1) ? fm1 : c;
    const v4f gl = (al - fm2) * wl;
    const v4f gr = (fp2 - ar) * wr;
    d1o = (fp1 - fm1) * wc;
    d2o = (gr - gl) * wc;
}

// ---------------- WMMA-based wave reduction (exact fp32) ----------------
__device__ __forceinline__ float wave_sum_wmma(float s) {
    v2f a; a[0] = s;    a[1] = 0.0f;
    v2f b; b[0] = 1.0f; b[1] = 1.0f;
    v8f c = {};
    v8f d = __builtin_amdgcn_wmma_f32_16x16x4_f32(
        false, a, false, b, (short)0, c, false, false);
    float r = d[0] + d[1] + d[2] + d[3] + d[4] + d[5] + d[6] + d[7];
    r += __shfl_xor(r, 16, 32);
    return r;   // all lanes hold the full 32-lane sum
}

// ---------------- Kernel A: elementwise losses ----------------
__global__ void loss_elem_kernel(const float* __restrict__ X1,
                                 const float* __restrict__ X,
                                 const float* __restrict__ Y,
                                 const float* __restrict__ C,
                                 const float* __restrict__ Xlast,
                                 float* __restrict__ partials) {
    float s1 = 0.f, s2 = 0.f, s3 = 0.f, s4 = 0.f;
    for (int k = blockIdx.x * blockDim.x + threadIdx.x; k < NCHUNK;
         k += gridDim.x * blockDim.x) {
        const int idx = k * 4;
        const v4f x1 = *(const v4f*)(X1 + idx);
        const v4f xx = *(const v4f*)(X + idx);
        const v4f yy = *(const v4f*)(Y + idx);
        const v4f cc = *(const v4f*)(C + idx);
        const int t0 = idx & (TDIM - 1);
        const float prev = (t0 == 0) ? Xlast[idx >> 4] : Y[idx - 1];
        const v4f e1 = x1 - yy;
        const v4f e2 = xx - yy;
        const v4f e4 = cc - yy;
        v4f e3; e3[0] = prev - yy[0]; e3[1] = yy[0] - yy[1];
        e3[2] = yy[1] - yy[2];        e3[3] = yy[2] - yy[3];
        const v4f q1 = e1 * e1, q2 = e2 * e2, q3 = e3 * e3, q4 = e4 * e4;
        s1 += q1[0] + q1[1] + q1[2] + q1[3];
        s2 += q2[0] + q2[1] + q2[2] + q2[3];
        s3 += q3[0] + q3[1] + q3[2] + q3[3];
        s4 += q4[0] + q4[1] + q4[2] + q4[3];
    }
    for (int off = 16; off; off >>= 1) {
        s1 += __shfl_xor(s1, off, 32);
        s2 += __shfl_xor(s2, off, 32);
        s3 += __shfl_xor(s3, off, 32);
        s4 += __shfl_xor(s4, off, 32);
    }
    __shared__ float red[8][4];
    const int lane = threadIdx.x & 31, wave = threadIdx.x >> 5;
    if (lane == 0) {
        red[wave][0] = s1; red[wave][1] = s2; red[wave][2] = s3; red[wave][3] = s4;
    }
    __syncthreads();
    if (threadIdx.x == 0) {
        float a0 = 0.f, a1 = 0.f, a2 = 0.f, a3 = 0.f;
        for (int w = 0; w < 8; ++w) {
            a0 += red[w][0]; a1 += red[w][1]; a2 += red[w][2]; a3 += red[w][3];
        }
        float* o = partials + blockIdx.x * 4;
        o[0] = a0; o[1] = a1; o[2] = a2; o[3] = a3;
    }
}

// ---------------- Kernel B: physics residual loss (branchless, float4 over t) ----------------
// V/F layout: (B, Y, X, Z, 3, T); P: (B, Y, X, Z, T); T innermost.
__global__ void phy_kernel(const float* __restrict__ V,
                           const float* __restrict__ P,
                           const float* __restrict__ F,
                           const float* __restrict__ ReP,
                           float* __restrict__ partials) {
    const float Re = ReP[0];
    const int VSZ = 3 * TDIM;       // 48
    const int VSX = SDIM * VSZ;     // 2304
    const int VSY = SDIM * VSX;     // 110592
    const int PSZ = TDIM;
    const int PSX = SDIM * PSZ;
    const int PSY = SDIM * PSX;
    v4f accv = {0.f, 0.f, 0.f, 0.f};

    for (int k = blockIdx.x * blockDim.x + threadIdx.x; k < NCHUNK;
         k += gridDim.x * blockDim.x) {
        const int t0 = (k & 3) * 4;
        int r = k >> 2;
        const int z = r % SDIM; r /= SDIM;
        const int x = r % SDIM; r /= SDIM;
        const int y = r % SDIM;
        const int b = r / SDIM;
        const int point = ((b * SDIM + y) * SDIM + x) * SDIM + z;
        const int vbase = point * VSZ + t0;
        const int pbase = point * PSZ + t0;

        const float* pVy = V + vbase;             // component 0
        const float* pVx = V + vbase + TDIM;      // component 1
        const float* pVz = V + vbase + 2 * TDIM;  // component 2
        const float* pP  = P + pbase;

        const v4f cy = *(const v4f*)pVy;
        const v4f cx = *(const v4f*)pVx;
        const v4f cz = *(const v4f*)pVz;

        const v4f vy_t = tderiv(pVy, t0, cy);
        const v4f vx_t = tderiv(pVx, t0, cx);
        const v4f vz_t = tderiv(pVz, t0, cz);

        v4f vy_y, vy_x, vy_z, vx_y, vx_x, vx_z, vz_y, vz_x, vz_z;
        v4f a, bb, cvec;
        sderiv12(pVy, y, VSY, cy, vy_y, a);
        sderiv12(pVy, x, VSX, cy, vy_x, bb);
        sderiv12(pVy, z, VSZ, cy, vy_z, cvec);
        const v4f lapy = a + bb + cvec;
        sderiv12(pVx, y, VSY, cx, vx_y, a);
        sderiv12(pVx, x, VSX, cx, vx_x, bb);
        sderiv12(pVx, z, VSZ, cx, vx_z, cvec);
        const v4f lapx = a + bb + cvec;
        sderiv12(pVz, y, VSY, cz, vz_y, a);
        sderiv12(pVz, x, VSX, cz, vz_x, bb);
        sderiv12(pVz, z, VSZ, cz, vz_z, cvec);
        const v4f lapz = a + bb + cvec;

        const v4f dPy = sderiv1(pP, y, PSY);
        const v4f dPx = sderiv1(pP, x, PSX);
        const v4f dPz = sderiv1(pP, z, PSZ);

        const v4f fy = *(const v4f*)(F + vbase);
        const v4f fx = *(const v4f*)(F + vbase + TDIM);
        const v4f fz = *(const v4f*)(F + vbase + 2 * TDIM);

        const v4f e1 = vy_t + (cx * vy_x + cy * vy_y + cz * vy_z + dPy) - Re * lapy + fy;
        const v4f e2 = vx_t + (cx * vx_x + cy * vx_y + cz * vx_z + dPx) - Re * lapx + fx;
        const v4f e3 = vz_t + (cx * vz_x + cy * vz_y + cz * vz_z + dPz) - Re * lapz + fz;
        const v4f e4 = vx_x + vy_y + vz_z;

        accv += e1 * e1 + e2 * e2 + e3 * e3 + e4 * e4;
    }
    const float acc = accv[0] + accv[1] + accv[2] + accv[3];

    // Matrix-unit wave reduction (all 32 lanes active; EXEC all-1s)
    const float wsum = wave_sum_wmma(acc);
    __shared__ float red[8];
    const int lane = threadIdx.x & 31, wave = threadIdx.x >> 5;
    if (lane == 0) red[wave] = wsum;
    __syncthreads();
    if (threadIdx.x == 0) {
        float s = 0.f;
        for (int w = 0; w < 8; ++w) s += red[w];
        partials[blockIdx.x] = s;
    }
}

// ---------------- Finalize: deterministic fixed-tree reduction ----------------
__global__ void finalize_kernel(const float* __restrict__ pa,
                                const float* __restrict__ pb,
                                float* __restrict__ out) {
    __shared__ float sd[256];
    const int tid = threadIdx.x;
    float sums[4];
    for (int c = 0; c < 4; ++c) {
        float s = pa[tid * 4 + c] + pa[(tid + 256) * 4 + c];
        sd[tid] = s;
        __syncthreads();
        for (int off = 128; off; off >>= 1) {
            if (tid < off) sd[tid] += sd[tid + off];
            __syncthreads();
        }
        sums[c] = sd[0];
        __syncthreads();
    }
    float sb = pb[tid] + pb[tid + 256] + pb[tid + 512] + pb[tid + 768];
    sd[tid] = sb;
    __syncthreads();
    for (int off = 128; off; off >>= 1) {
        if (tid < off) sd[tid] += sd[tid + off];
        __syncthreads();
    }
    if (tid == 0) {
        const float invN = 1.0f / (float)NTOT;
        const float main1 = sums[0] * invN;
        const float main2 = sums[1] * invN;
        const float t1    = sums[2] * invN;
        const float t2    = sums[3] * invN;
        const float phy   = sd[0] * invN;
        out[0] = main1;
        out[1] = main2;
        out[2] = phy;
        out[3] = (t1 < t2) ? (t2 - t1) : 0.0f;
    }
}

extern "C" void kernel_launch(void* const* d_in, const int* in_sizes, int n_in,
                              void* d_out, int out_size, void* d_ws, size_t ws_size,
                              hipStream_t stream) {
    const float* C  = (const float*)d_in[0];
    const float* V  = (const float*)d_in[1];
    const float* P  = (const float*)d_in[2];
    const float* X  = (const float*)d_in[3];
    const float* X1 = (const float*)d_in[4];
    const float* F  = (const float*)d_in[5];
    const float* Re = (const float*)d_in[6];
    const float* Xl = (const float*)d_in[7];
    const float* Y  = (const float*)d_in[8];
    // masks d_in[9..11] are unused by the reference outputs.

    float* ws = (float*)d_ws;
    float* pa = ws;                 // GRID_A * 4 floats
    float* pb = ws + GRID_A * 4;    // GRID_B floats

    loss_elem_kernel<<<GRID_A, 256, 0, stream>>>(X1, X, Y, C, Xl, pa);
    phy_kernel<<<GRID_B, 256, 0, stream>>>(V, P, F, Re, pb);
    finalize_kernel<<<1, 256, 0, stream>>>(pa, pb, (float*)d_out);
}